// Convolution_52862457479225
// MI455X (gfx1250) — compile-verified
//
#include <hip/hip_runtime.h>
#include <hip/hip_bf16.h>

// Problem constants from setup_inputs(): B=2, P=256, C=J=I=32, NB=16.
#define BATCH   2
#define NPTS    256
#define NCH     32
#define NBASIS  16
#define KDIM    (NBASIS * NCH)   // 512 = fused (n,j) contraction dim for GEMM-2
#define GAMMA_F 10.0f
#define EPS_F   1e-9f

typedef float v2f __attribute__((ext_vector_type(2)));
typedef float v8f __attribute__((ext_vector_type(8)));

// ---------------------------------------------------------------------------
// Workspace layout (floats):
//   [0, 131072)            : D[z][a][b]            pairwise distances (512 KB)
//   [131072, 393216)       : S[z][a][n*32+j]       stage-1 result     (1 MB)
//   [393216, 409600)       : W2[(n*32+j)][i]       kernel_w transposed * 1/sqrt(P)
// ---------------------------------------------------------------------------
#define OFF_D   0
#define OFF_S   (BATCH * NPTS * NPTS)                  // 131072
#define OFF_W2  (OFF_S + BATCH * NPTS * KDIM)          // 393216

// Kernel 1: pairwise safe-norm distances d[z,a,b] = sqrt(|g_a-g_b|^2 + eps).
__global__ void cf_dist_kernel(const float* __restrict__ geom,
                               float* __restrict__ dmat) {
    int idx = blockIdx.x * blockDim.x + threadIdx.x;   // [0, 131072)
    int b = idx & (NPTS - 1);
    int a = (idx >> 8) & (NPTS - 1);
    int z = idx >> 16;
    const float* ga = geom + (z * NPTS + a) * 3;
    const float* gb = geom + (z * NPTS + b) * 3;
    float dx = gb[0] - ga[0];
    float dy = gb[1] - ga[1];
    float dz = gb[2] - ga[2];
    dmat[idx] = sqrtf(dx * dx + dy * dy + dz * dz + EPS_F);
}

// Kernel 2: W2[(n*32+j)*32 + i] = kernel_w[n][i][j] * rsqrt(n_norm).
__global__ void cf_w2_kernel(const float* __restrict__ kw,
                             const int* __restrict__ n_norm,
                             float* __restrict__ w2) {
    int idx = blockIdx.x * blockDim.x + threadIdx.x;   // [0, 16384)
    float scale = rsqrtf((float)n_norm[0]);
    int i  = idx & 31;
    int nj = idx >> 5;
    int j  = nj & 31;
    int n  = nj >> 5;
    w2[idx] = kw[(n * NCH + i) * NCH + j] * scale;
}

// Kernel 3 (WMMA GEMM-1, RBF fused into A-fragment):
//   S_n = RBF_n(256x256) @ feat(256x32)  for each (z, n).
// One wave per 16x32 output tile (two 16x16 accumulators sharing one A
// fragment): 2*16*16 = 512 waves, K=256 -> 64 k-steps, 128 WMMAs per wave.
// The RBF (pk_add + pk_mul*3 + 2x v_exp_f32) is evaluated ONCE per k-step and
// feeds both WMMAs, halving the transcendental VALU load per matrix op.
__global__ void cf_gemm1_kernel(const float* __restrict__ dmat,
                                const float* __restrict__ feat,
                                const float* __restrict__ centers,
                                float* __restrict__ S) {
    const int wave  = blockIdx.x * (blockDim.x >> 5) + (threadIdx.x >> 5);
    const int lane  = threadIdx.x & 31;
    const int at    = wave & 15;         // a-tile
    const int n     = (wave >> 4) & 15;  // radial basis index
    const int z     = wave >> 8;         // batch
    const int row0  = at * 16;
    const int m     = lane & 15;         // row (A) / col (B) within tile
    const int khalf = lane >> 4;         // 0: K=0,1   1: K=2,3

    const float  cn   = centers[n];
    const float* drow = dmat + (z * NPTS + row0 + m) * NPTS; // A row for this lane
    const float* fc0  = feat + z * NPTS * NCH + m;           // B cols 0..15
    const float* fc1  = fc0 + 16;                            // B cols 16..31

    v8f c0 = {};
    v8f c1 = {};
    for (int k0 = 0; k0 < NPTS; k0 += 4) {
        int b0 = k0 + khalf * 2;
        // Shared A fragment: fused RBF evaluation
        float t0 = drow[b0] - cn;
        float t1 = drow[b0 + 1] - cn;
        v2f av;
        av.x = __expf(-GAMMA_F * t0 * t0);
        av.y = __expf(-GAMMA_F * t1 * t1);
        // B fragments for j-tiles 0 and 1
        v2f bv0, bv1;
        bv0.x = fc0[b0 * NCH];
        bv0.y = fc0[(b0 + 1) * NCH];
        bv1.x = fc1[b0 * NCH];
        bv1.y = fc1[(b0 + 1) * NCH];
        c0 = __builtin_amdgcn_wmma_f32_16x16x4_f32(
                false, av, false, bv0, (short)0, c0, false, false);
        c1 = __builtin_amdgcn_wmma_f32_16x16x4_f32(
                false, av, false, bv1, (short)0, c1, false, false);
    }
    // D layout: VGPR i -> row = row0 + i + 8*khalf, col = m (tile 0) / 16+m.
    // Store into S[z][a][n*32 + j] so GEMM-2 sees contiguous 512-wide rows.
#pragma unroll
    for (int i = 0; i < 8; ++i) {
        int row = row0 + i + khalf * 8;
        float* srow = S + (z * NPTS + row) * KDIM + n * NCH;
        srow[m]      = c0[i];
        srow[16 + m] = c1[i];
    }
}

// Kernel 4 (WMMA GEMM-2): out[z] (256x32) = S[z] (256x512) @ W2 (512x32).
// One wave per 16x32 output tile (both i-tiles, shared A loads): 32 waves,
// K=512 -> 128 k-steps, 256 WMMAs per wave.
__global__ void cf_gemm2_kernel(const float* __restrict__ S,
                                const float* __restrict__ w2,
                                float* __restrict__ out) {
    const int wave  = blockIdx.x * (blockDim.x >> 5) + (threadIdx.x >> 5);
    const int lane  = threadIdx.x & 31;
    const int at    = wave & 15;         // a-tile
    const int z     = wave >> 4;         // batch
    const int row0  = at * 16;
    const int m     = lane & 15;
    const int khalf = lane >> 4;

    const float* srow = S + (z * NPTS + row0 + m) * KDIM;
    const float* wc0  = w2 + m;          // B cols 0..15
    const float* wc1  = wc0 + 16;        // B cols 16..31

    v8f c0 = {};
    v8f c1 = {};
    for (int k0 = 0; k0 < KDIM; k0 += 4) {
        int kk = k0 + khalf * 2;
        v2f av;                          // shared A fragment (b64 load)
        av.x = srow[kk];
        av.y = srow[kk + 1];
        v2f bv0, bv1;
        bv0.x = wc0[kk * NCH];
        bv0.y = wc0[(kk + 1) * NCH];
        bv1.x = wc1[kk * NCH];
        bv1.y = wc1[(kk + 1) * NCH];
        c0 = __builtin_amdgcn_wmma_f32_16x16x4_f32(
                false, av, false, bv0, (short)0, c0, false, false);
        c1 = __builtin_amdgcn_wmma_f32_16x16x4_f32(
                false, av, false, bv1, (short)0, c1, false, false);
    }
#pragma unroll
    for (int i = 0; i < 8; ++i) {
        int row = row0 + i + khalf * 8;
        float* orow = out + (z * NPTS + row) * NCH;
        orow[m]      = c0[i];
        orow[16 + m] = c1[i];
    }
}

extern "C" void kernel_launch(void* const* d_in, const int* in_sizes, int n_in,
                              void* d_out, int out_size, void* d_ws, size_t ws_size,
                              hipStream_t stream) {
    const float* feat    = (const float*)d_in[0];  // [2,256,32]
    const float* geom    = (const float*)d_in[1];  // [2,256,3]
    const float* centers = (const float*)d_in[2];  // [16]
    const float* kw      = (const float*)d_in[3];  // [16,32,32]
    const int*   n_norm  = (const int*)d_in[4];    // scalar 256

    float* ws   = (float*)d_ws;
    float* dmat = ws + OFF_D;
    float* S    = ws + OFF_S;
    float* w2   = ws + OFF_W2;
    float* out  = (float*)d_out;                   // [2,256,32]

    // Stage 0: distances (131072 elems) and scaled transposed weights (16384).
    cf_dist_kernel<<<BATCH * NPTS * NPTS / 256, 256, 0, stream>>>(geom, dmat);
    cf_w2_kernel<<<KDIM * NCH / 256, 256, 0, stream>>>(kw, n_norm, w2);

    // Stage 1: 512 tile-waves, 4 waves (128 threads) per block.
    cf_gemm1_kernel<<<128, 128, 0, stream>>>(dmat, feat, centers, S);

    // Stage 2: 32 tile-waves.
    cf_gemm2_kernel<<<8, 128, 0, stream>>>(S, w2, out);
}